// HarmonicGraphEncoder_70463233458547
// MI455X (gfx1250) — compile-verified
//
#include <hip/hip_runtime.h>

#define HID 128
#define LDS_STRIDE 132   // 128 + 4 pad -> conflict-free strided LDS reads

typedef __attribute__((ext_vector_type(2))) float v2f;
typedef __attribute__((ext_vector_type(8))) float v8f;

// ---------------------------------------------------------------------------
// h = x @ W_in + b_in        (N x 24) @ (24 x 128)  -- tiny, memory bound
// thread = (node, group of 4 hidden channels)
// ---------------------------------------------------------------------------
__global__ __launch_bounds__(256) void gine_encode(
    const float* __restrict__ x, const float* __restrict__ Win,
    const float* __restrict__ bin, float* __restrict__ h,
    int n_nodes, int node_dim)
{
    long tid  = (long)blockIdx.x * 256 + threadIdx.x;
    int  node = (int)(tid >> 5);
    int  c    = ((int)tid & 31) << 2;
    if (node >= n_nodes) return;

    const float* xr = x + (long)node * node_dim;
    float4 acc = make_float4(bin[c], bin[c + 1], bin[c + 2], bin[c + 3]);
    for (int k = 0; k < node_dim; ++k) {
        float xv = xr[k];
        const float* w = Win + (long)k * HID + c;
        acc.x += xv * w[0];
        acc.y += xv * w[1];
        acc.z += xv * w[2];
        acc.w += xv * w[3];
    }
    *(float4*)(h + (long)node * HID + c) = acc;
}

// ---------------------------------------------------------------------------
// Edge scatter: agg[dst] += relu(h[src] + edge_attr @ W_e + b_e)
// One wave per edge: lane l owns hidden channels [4l, 4l+4).
// Gather and atomic streams are fully coalesced 512B/wave; h & agg are
// L2-resident (51.2 MB each vs 192 MB L2). relu==0 messages skip the atomic
// (adding +0 is exact since agg starts at +0 and only receives non-negatives).
// ---------------------------------------------------------------------------
__global__ __launch_bounds__(256) void gine_scatter(
    const float* __restrict__ h, const int* __restrict__ ei,
    const float* __restrict__ ea, const float* __restrict__ We,
    const float* __restrict__ be, float* __restrict__ agg, int n_edges)
{
    long gtid = (long)blockIdx.x * 256 + threadIdx.x;
    int  e    = (int)(gtid >> 5);
    if (e >= n_edges) return;
    int lane = threadIdx.x & 31;

    int src = ei[e];
    int dst = ei[n_edges + e];
    float2 a = *(const float2*)(ea + (long)e * 2);

    int c = lane << 2;
    float4 hv = *(const float4*)(h + (long)src * HID + c);
    float4 w0 = *(const float4*)(We + c);
    float4 w1 = *(const float4*)(We + HID + c);
    float4 bv = *(const float4*)(be + c);

    float4 m;
    m.x = fmaxf(hv.x + a.x * w0.x + a.y * w1.x + bv.x, 0.0f);
    m.y = fmaxf(hv.y + a.x * w0.y + a.y * w1.y + bv.y, 0.0f);
    m.z = fmaxf(hv.z + a.x * w0.z + a.y * w1.z + bv.z, 0.0f);
    m.w = fmaxf(hv.w + a.x * w0.w + a.y * w1.w + bv.w, 0.0f);

    float* ap = agg + (long)dst * HID + c;
    if (m.x != 0.0f) unsafeAtomicAdd(ap + 0, m.x);
    if (m.y != 0.0f) unsafeAtomicAdd(ap + 1, m.y);
    if (m.z != 0.0f) unsafeAtomicAdd(ap + 2, m.z);
    if (m.w != 0.0f) unsafeAtomicAdd(ap + 3, m.w);
}

// ---------------------------------------------------------------------------
// Fused node update MLP (shared by both convs):
//   z = h + agg ; t = relu(z@W1 + b1) ; out = t@W2 + b2 (optional relu)
// Block = 16 node rows x full 128 cols, 8 waves (wave w -> col tile 16w..16w+15).
// Both GEMMs run on v_wmma_f32_16x16x4_f32 (exact fp32), K-loop of 32 steps.
// A fragments come from LDS (stride-132 rows -> conflict-free b64 loads),
// B fragments from global W (64KB, L0/L2 hot). No divergence -> EXEC all ones.
// In-place h update is safe: each block reads only its own rows before writing.
// ---------------------------------------------------------------------------
__global__ __launch_bounds__(256) void gine_mlp(
    const float* __restrict__ h, const float* __restrict__ agg,
    const float* __restrict__ W1, const float* __restrict__ b1,
    const float* __restrict__ W2, const float* __restrict__ b2,
    float* out, int n_nodes, int relu_out)
{
    __shared__ float ldsZ[16 * LDS_STRIDE];
    __shared__ float ldsT[16 * LDS_STRIDE];

    int m0  = blockIdx.x * 16;
    int tid = threadIdx.x;

    // ---- stage z = h + agg : 16 rows x 128 cols, float4 granularity ----
    for (int i = tid; i < 16 * (HID / 4); i += 256) {
        int r  = i >> 5;        // row within tile
        int c4 = i & 31;        // float4 column
        int row = m0 + r;
        float4 zv = make_float4(0.f, 0.f, 0.f, 0.f);
        if (row < n_nodes) {
            float4 hv = *(const float4*)(h   + (long)row * HID + c4 * 4);
            float4 av = *(const float4*)(agg + (long)row * HID + c4 * 4);
            zv = make_float4(hv.x + av.x, hv.y + av.y, hv.z + av.z, hv.w + av.w);
        }
        *(float4*)(&ldsZ[r * LDS_STRIDE + c4 * 4]) = zv;
    }
    __syncthreads();

    int wave  = tid >> 5;
    int lane  = tid & 31;
    int col0  = wave << 4;
    int lrow  = lane & 15;          // A row / C col within tile
    int khalf = (lane >> 4) << 1;   // 0 or 2: which K pair this half-wave owns
    int col   = col0 + lrow;        // B/C column in [0,128)
    int mbase = (lane >> 4) << 3;   // C rows: 0..7 or 8..15

    // ---- GEMM1: t = relu(z @ W1 + b1) ----
    v8f acc1 = {0.f, 0.f, 0.f, 0.f, 0.f, 0.f, 0.f, 0.f};
#pragma unroll
    for (int k = 0; k < HID; k += 4) {
        v2f a = *(const v2f*)(&ldsZ[lrow * LDS_STRIDE + k + khalf]);
        v2f b;
        b[0] = W1[(k + khalf) * HID + col];
        b[1] = W1[(k + khalf + 1) * HID + col];
        acc1 = __builtin_amdgcn_wmma_f32_16x16x4_f32(
            false, a, false, b, (short)0, acc1, false, false);
    }
    {
        float bias = b1[col];
#pragma unroll
        for (int r = 0; r < 8; ++r) {
            float v = acc1[r] + bias;
            ldsT[(mbase + r) * LDS_STRIDE + col] = fmaxf(v, 0.0f);
        }
    }
    __syncthreads();

    // ---- GEMM2: out = t @ W2 + b2 ----
    v8f acc2 = {0.f, 0.f, 0.f, 0.f, 0.f, 0.f, 0.f, 0.f};
#pragma unroll
    for (int k = 0; k < HID; k += 4) {
        v2f a = *(const v2f*)(&ldsT[lrow * LDS_STRIDE + k + khalf]);
        v2f b;
        b[0] = W2[(k + khalf) * HID + col];
        b[1] = W2[(k + khalf + 1) * HID + col];
        acc2 = __builtin_amdgcn_wmma_f32_16x16x4_f32(
            false, a, false, b, (short)0, acc2, false, false);
    }
    {
        float bias = b2[col];
#pragma unroll
        for (int r = 0; r < 8; ++r) {
            int row = m0 + mbase + r;
            if (row < n_nodes) {
                float v = acc2[r] + bias;
                if (relu_out) v = fmaxf(v, 0.0f);
                out[(long)row * HID + col] = v;
            }
        }
    }
}

// ---------------------------------------------------------------------------
extern "C" void kernel_launch(void* const* d_in, const int* in_sizes, int n_in,
                              void* d_out, int out_size, void* d_ws, size_t ws_size,
                              hipStream_t stream)
{
    const float* x   = (const float*)d_in[0];
    const int*   ei  = (const int*)  d_in[1];   // edge_index [2, E]
    const float* ea  = (const float*)d_in[2];   // edge_attr  [E, 2]
    const float* Win = (const float*)d_in[3];
    const float* bin = (const float*)d_in[4];
    const float* We  = (const float*)d_in[5];
    const float* be  = (const float*)d_in[6];
    const float* W1  = (const float*)d_in[7];
    const float* b1  = (const float*)d_in[8];
    const float* W2  = (const float*)d_in[9];
    const float* b2  = (const float*)d_in[10];

    int node_dim = in_sizes[3] / HID;          // 24
    int n_nodes  = in_sizes[0] / node_dim;     // 100000
    int n_edges  = in_sizes[2] / 2;            // 1600000

    float* h   = (float*)d_out;                // [N,128]; final output lives here
    float* agg = (float*)d_ws;                 // [N,128] scatter accumulator

    size_t aggBytes = (size_t)n_nodes * HID * sizeof(float);

    // encode: h = x @ W_in + b_in
    {
        long th = (long)n_nodes * 32;
        int blocks = (int)((th + 255) / 256);
        gine_encode<<<blocks, 256, 0, stream>>>(x, Win, bin, h, n_nodes, node_dim);
    }

    int mlpBlocks = (n_nodes + 15) / 16;
    long sth = (long)n_edges * 32;
    int sBlocks = (int)((sth + 255) / 256);

    for (int conv = 0; conv < 2; ++conv) {
        hipMemsetAsync(agg, 0, aggBytes, stream);
        gine_scatter<<<sBlocks, 256, 0, stream>>>(h, ei, ea, We, be, agg, n_edges);
        gine_mlp<<<mlpBlocks, 256, 0, stream>>>(h, agg, W1, b1, W2, b2, h,
                                                n_nodes, conv == 0 ? 1 : 0);
    }
}